// GCN_53472342835252
// MI455X (gfx1250) — compile-verified
//
#include <hip/hip_runtime.h>
#include <stdint.h>

typedef float v2f_t __attribute__((ext_vector_type(2)));
typedef float v8f_t __attribute__((ext_vector_type(8)));

#define GCN_D 128

// ---------------- degree / normalization ----------------
__global__ void gcn_deg_init(float* __restrict__ deg, int N) {
  int n = blockIdx.x * blockDim.x + threadIdx.x;
  if (n < N) deg[n] = 1.0f;  // self-loop contribution
}

__global__ void gcn_deg_accum(const long long* __restrict__ ei,
                              float* __restrict__ deg, int E) {
  int e = blockIdx.x * blockDim.x + threadIdx.x;
  if (e < E) {
    int c = (int)ei[(size_t)E + e];  // col
    unsafeAtomicAdd(&deg[c], 1.0f);
  }
}

__global__ void gcn_deg_rsqrt(float* __restrict__ deg, int N) {
  int n = blockIdx.x * blockDim.x + threadIdx.x;
  if (n < N) deg[n] = rsqrtf(deg[n]);  // deg >= 1 always (self loops)
}

// ---------------- GEMM: H[N,128] = X[N,128] * W[128,128] (f32 WMMA) ---------
// One wave computes 16 rows x all 128 cols (8 column tiles of 16).
// W is staged in LDS pre-swizzled as K-pairs: sWp[k/2][col] = {W[k][col], W[k+1][col]}
// so each B fragment is a single contiguous 8-byte ds_load_b64 into an even
// VGPR pair, with all loop offsets as immediates off one lane base address.
// A frag (16x4 f32, 2 VGPR): lanes 0-15 -> K={k0,k0+1}, lanes 16-31 -> K={k0+2,k0+3}, M=lane&15
// C/D (16x16 f32, 8 VGPR): element v -> row m0+v+8*(lane>>4), col = nt*16+(lane&15)
__global__ void __launch_bounds__(256) gcn_gemm_wmma(
    const float* __restrict__ X, const float* __restrict__ W,
    float* __restrict__ H, int N) {
  __shared__ float2 sWp[(GCN_D / 2) * GCN_D];  // 64 KB
  // cooperative staging: interleave K-pairs, coalesced float4 row reads
  for (int i = threadIdx.x; i < (GCN_D / 2) * (GCN_D / 4); i += blockDim.x) {
    int k2 = i >> 5;           // pair row 0..63
    int c4 = (i & 31) * 4;     // col group
    float4 r0 = *(const float4*)(W + (size_t)(2 * k2) * GCN_D + c4);
    float4 r1 = *(const float4*)(W + (size_t)(2 * k2 + 1) * GCN_D + c4);
    float4* dst = (float4*)(sWp + (size_t)k2 * GCN_D + c4);
    dst[0] = make_float4(r0.x, r1.x, r0.y, r1.y);
    dst[1] = make_float4(r0.z, r1.z, r0.w, r1.w);
  }
  __syncthreads();

  int lane = threadIdx.x & 31;
  int wave = blockIdx.x * (blockDim.x >> 5) + (threadIdx.x >> 5);
  int m0 = wave * 16;
  if (m0 < N) {
    int half = lane >> 4;   // 0 or 1
    int l15  = lane & 15;
    int mr = m0 + l15; if (mr > N - 1) mr = N - 1;   // clamp, keep EXEC uniform
    const float* xrow = X + (size_t)mr * GCN_D;
    const float2* bbase = sWp + half * GCN_D + l15;  // lane base; rest immediates

    v8f_t acc[8] = {};
    for (int k0 = 0; k0 < GCN_D; k0 += 4) {
      int kk = k0 + 2 * half;
      v2f_t a = *(const v2f_t*)(xrow + kk);          // global_load_b64
      int kb = (k0 >> 1) * GCN_D;
#pragma unroll
      for (int nt = 0; nt < 8; ++nt) {
        v2f_t b = *(const v2f_t*)(bbase + kb + nt * 16);  // ds_load_b64, imm offset
        acc[nt] = __builtin_amdgcn_wmma_f32_16x16x4_f32(
            false, a, false, b, (short)0, acc[nt], false, false);
      }
    }
#pragma unroll
    for (int nt = 0; nt < 8; ++nt) {
#pragma unroll
      for (int v = 0; v < 8; ++v) {
        int row = m0 + v + 8 * half;
        if (row < N)
          H[(size_t)row * GCN_D + nt * 16 + l15] = acc[nt][v];
      }
    }
  }
}

// ---------------- out[n,:] = H[n,:] * dinv[n]^2 + b  (self-loop + bias) ----
__global__ void gcn_init_out(const float* __restrict__ H,
                             const float* __restrict__ dinv,
                             const float* __restrict__ b,
                             float* __restrict__ out, int N) {
  int t = blockIdx.x * blockDim.x + threadIdx.x;
  int n = t >> 5, lane = t & 31;  // 32 lanes x float4 = 128 features
  if (n >= N) return;
  float s = dinv[n]; s *= s;
  float4 h  = ((const float4*)(H + (size_t)n * GCN_D))[lane];
  float4 bb = ((const float4*)b)[lane];
  float4 o = make_float4(h.x * s + bb.x, h.y * s + bb.y,
                         h.z * s + bb.z, h.w * s + bb.w);
  ((float4*)(out + (size_t)n * GCN_D))[lane] = o;
}

// ---------------- edge scatter: out[col,:] += H[row,:] * dinv[r]*dinv[c] ---
__global__ void gcn_scatter(const float* __restrict__ H,
                            const float* __restrict__ dinv,
                            const long long* __restrict__ ei,
                            float* __restrict__ out, int E) {
  int t = blockIdx.x * blockDim.x + threadIdx.x;
  int e = t >> 5, lane = t & 31;  // one wave per edge
  if (e >= E) return;
  int r = (int)ei[e];
  int c = (int)ei[(size_t)E + e];
  float w = dinv[r] * dinv[c];
  float4 h = ((const float4*)(H + (size_t)r * GCN_D))[lane];
  float* o = out + (size_t)c * GCN_D + lane * 4;
  unsafeAtomicAdd(o + 0, h.x * w);
  unsafeAtomicAdd(o + 1, h.y * w);
  unsafeAtomicAdd(o + 2, h.z * w);
  unsafeAtomicAdd(o + 3, h.w * w);
}

// ---------------- in-place ReLU -------------------------------------------
__global__ void gcn_relu(float* __restrict__ x, int n) {  // n multiple of 4
  int i = blockIdx.x * blockDim.x + threadIdx.x;
  if (i * 4 < n) {
    float4* p = (float4*)x + i;
    float4 v = *p;
    v.x = fmaxf(v.x, 0.f); v.y = fmaxf(v.y, 0.f);
    v.z = fmaxf(v.z, 0.f); v.w = fmaxf(v.w, 0.f);
    *p = v;
  }
}

extern "C" void kernel_launch(void* const* d_in, const int* in_sizes, int n_in,
                              void* d_out, int out_size, void* d_ws, size_t ws_size,
                              hipStream_t stream) {
  const float*     X  = (const float*)d_in[0];     // [N,128] f32
  const long long* ei = (const long long*)d_in[1]; // [2,E] int64
  const float*     W1 = (const float*)d_in[2];
  const float*     b1 = (const float*)d_in[3];
  const float*     W2 = (const float*)d_in[4];
  const float*     b2 = (const float*)d_in[5];
  int N = in_sizes[0] / GCN_D;
  int E = in_sizes[1] / 2;

  // workspace: dinv [N] | H [N*128] | x1 [N*128]
  float* dinv = (float*)d_ws;
  size_t off = ((size_t)N * sizeof(float) + 255) & ~(size_t)255;
  float* H   = (float*)((char*)d_ws + off);
  float* x1  = H + (size_t)N * GCN_D;
  float* out = (float*)d_out;

  const int T = 256;
  int nodeB1 = (N + T - 1) / T;
  int edgeB1 = (E + T - 1) / T;
  int gemmB  = (((N + 15) / 16) + 7) / 8;              // 8 waves/block
  int nodeB32 = (int)(((long long)N * 32 + T - 1) / T);
  int edgeB32 = (int)(((long long)E * 32 + T - 1) / T);
  int elts = N * GCN_D;
  int reluB = (elts / 4 + T - 1) / T;

  // normalization coefficients
  gcn_deg_init <<<nodeB1, T, 0, stream>>>(dinv, N);
  gcn_deg_accum<<<edgeB1, T, 0, stream>>>(ei, dinv, E);
  gcn_deg_rsqrt<<<nodeB1, T, 0, stream>>>(dinv, N);

  // layer 1
  gcn_gemm_wmma<<<gemmB,   T, 0, stream>>>(X, W1, H, N);
  gcn_init_out <<<nodeB32, T, 0, stream>>>(H, dinv, b1, x1, N);
  gcn_scatter  <<<edgeB32, T, 0, stream>>>(H, dinv, ei, x1, E);
  gcn_relu     <<<reluB,   T, 0, stream>>>(x1, elts);

  // layer 2 (scatter straight into d_out)
  gcn_gemm_wmma<<<gemmB,   T, 0, stream>>>(x1, W2, H, N);
  gcn_init_out <<<nodeB32, T, 0, stream>>>(H, dinv, b2, out, N);
  gcn_scatter  <<<edgeB32, T, 0, stream>>>(H, dinv, ei, out, E);
  gcn_relu     <<<reluB,   T, 0, stream>>>(out, elts);
}